// DifficultyGNN_59562606460951
// MI455X (gfx1250) — compile-verified
//
#include <hip/hip_runtime.h>

typedef __attribute__((ext_vector_type(2))) float v2f;
typedef __attribute__((ext_vector_type(8))) float v8f;

#define N_NODES  100000
#define N_EDGES  1600000
#define IN_DIM   128
#define HID      64
#define HID2     32
#define N_GRAPHS 128
#define OUT_DIM  3

static __device__ __forceinline__ void atomAddF(float* p, float v) {
    // native global_atomic_add_f32 (no CAS loop)
    unsafeAtomicAdd(p, v);
}

// ---------- degree / norm precompute ----------
__global__ void k_fill1(float* p, int n) {
    int i = blockIdx.x * blockDim.x + threadIdx.x;
    if (i < n) p[i] = 1.0f;                       // self-loop contributes 1 to deg
}

__global__ void k_deg(const int* __restrict__ dst, float* deg, int e) {
    int i = blockIdx.x * blockDim.x + threadIdx.x;
    if (i < e) atomAddF(&deg[dst[i]], 1.0f);
}

__global__ void k_rsqrt(float* d, int n) {
    int i = blockIdx.x * blockDim.x + threadIdx.x;
    if (i < n) { float v = d[i]; d[i] = (v > 0.0f) ? rsqrtf(v) : 0.0f; }
}

__global__ void k_norm(const int* __restrict__ src, const int* __restrict__ dst,
                       const float* __restrict__ dinv, float* __restrict__ nrm, int e) {
    int i = blockIdx.x * blockDim.x + threadIdx.x;
    if (i < e) nrm[i] = dinv[src[i]] * dinv[dst[i]];
}

// ---------- fp32 WMMA GEMM: C[N,M] = A[N,K] @ B[K,M] ----------
// grid.x = N/16 row tiles; blockDim = (32, M/16): one wave per 16x16 output tile.
__global__ void k_gemm_wmma(const float* __restrict__ A, const float* __restrict__ B,
                            float* __restrict__ C, int K, int M) {
    const int row0 = blockIdx.x * 16;
    const int col0 = threadIdx.y * 16;
    const int lane = threadIdx.x;
    const int half = lane >> 4;     // 0: lanes 0-15, 1: lanes 16-31
    const int mn   = lane & 15;     // A-row / B-col / C-col index within tile

    const float* Arow = A + (size_t)(row0 + mn) * K;
    v8f acc = {};
    for (int k0 = 0; k0 < K; k0 += 4) {
        // A 16x4 frag: lane holds A[mn][k0+2*half], A[mn][k0+2*half+1] (contiguous)
        v2f a = *(const v2f*)(Arow + k0 + 2 * half);
        // B 4x16 frag: VGPR0 = rows K=0/2, VGPR1 = rows K=1/3, col = mn
        v2f b;
        b.x = B[(size_t)(k0 + 2 * half)     * M + col0 + mn];
        b.y = B[(size_t)(k0 + 2 * half + 1) * M + col0 + mn];
        acc = __builtin_amdgcn_wmma_f32_16x16x4_f32(false, a, false, b,
                                                    (short)0, acc, false, false);
    }
    // C 16x16 f32: VGPR r -> row (r + 8*half), col mn
    float* Crow = C + (size_t)(row0 + 8 * half) * M + col0 + mn;
#pragma unroll
    for (int r = 0; r < 8; ++r) Crow[(size_t)r * M] = acc[r];
}

// ---------- aggregation ----------
// agg = h * dinv^2  (self-loop term; also fully initializes agg, no memset needed)
__global__ void k_selfloop(const float* __restrict__ h, const float* __restrict__ dinv,
                           float* __restrict__ agg, int M) {
    int chunks = M >> 2;
    long long tid = (long long)blockIdx.x * blockDim.x + threadIdx.x;
    long long total = (long long)N_NODES * chunks;
    if (tid >= total) return;
    int node = (int)(tid / chunks);
    int c    = (int)(tid % chunks);
    float w = dinv[node]; w *= w;
    float4 hv = *(const float4*)(h + (size_t)node * M + c * 4);
    float4 o; o.x = hv.x * w; o.y = hv.y * w; o.z = hv.z * w; o.w = hv.w * w;
    *(float4*)(agg + (size_t)node * M + c * 4) = o;
}

// agg[dst] += h[src] * norm[e] ; 16 (or 8) consecutive lanes per edge -> coalesced gathers
__global__ void k_scatter(const float* __restrict__ h, const int* __restrict__ src,
                          const int* __restrict__ dst, const float* __restrict__ nrm,
                          float* __restrict__ agg, int M) {
    int chunks = M >> 2;
    long long tid = (long long)blockIdx.x * blockDim.x + threadIdx.x;
    long long total = (long long)N_EDGES * chunks;
    if (tid >= total) return;
    int e = (int)(tid / chunks);
    int c = (int)(tid % chunks);
    int s = src[e], d = dst[e];
    float w = nrm[e];
    float4 hv = *(const float4*)(h + (size_t)s * M + c * 4);
    float* ap = agg + (size_t)d * M + c * 4;
    atomAddF(ap + 0, hv.x * w);
    atomAddF(ap + 1, hv.y * w);
    atomAddF(ap + 2, hv.z * w);
    atomAddF(ap + 3, hv.w * w);
}

__global__ void k_bias_relu(const float* __restrict__ agg, const float* __restrict__ b,
                            float* __restrict__ out, int M) {
    long long i = (long long)blockIdx.x * blockDim.x + threadIdx.x;
    if (i >= (long long)N_NODES * M) return;
    int f = (int)(i % M);
    out[i] = fmaxf(agg[i] + b[f], 0.0f);
}

// ---------- pooling + classifier ----------
__global__ void k_count(const int* __restrict__ batch, float* counts, int n) {
    int i = blockIdx.x * blockDim.x + threadIdx.x;
    if (i < n) atomAddF(&counts[batch[i]], 1.0f);
}

__global__ void k_pool(const float* __restrict__ h, const int* __restrict__ batch,
                       float* __restrict__ pooled) {
    const int chunks = HID2 >> 2;  // 8
    long long tid = (long long)blockIdx.x * blockDim.x + threadIdx.x;
    long long total = (long long)N_NODES * chunks;
    if (tid >= total) return;
    int node = (int)(tid / chunks);
    int c    = (int)(tid % chunks);
    int g = batch[node];
    float4 hv = *(const float4*)(h + (size_t)node * HID2 + c * 4);
    float* pp = pooled + (size_t)g * HID2 + c * 4;
    atomAddF(pp + 0, hv.x);
    atomAddF(pp + 1, hv.y);
    atomAddF(pp + 2, hv.z);
    atomAddF(pp + 3, hv.w);
}

__global__ void k_classifier(const float* __restrict__ pooled, const float* __restrict__ counts,
                             const float* __restrict__ Wc, const float* __restrict__ bc,
                             float* __restrict__ out) {
    int g = threadIdx.x;
    if (g >= N_GRAPHS) return;
    float inv = 1.0f / fmaxf(counts[g], 1.0f);
    float lg[OUT_DIM];
#pragma unroll
    for (int c = 0; c < OUT_DIM; ++c) {
        float s = bc[c];
        for (int f = 0; f < HID2; ++f)
            s += pooled[g * HID2 + f] * inv * Wc[f * OUT_DIM + c];
        lg[c] = s;
    }
    float m = fmaxf(lg[0], fmaxf(lg[1], lg[2]));
    float lse = logf(expf(lg[0] - m) + expf(lg[1] - m) + expf(lg[2] - m));
#pragma unroll
    for (int c = 0; c < OUT_DIM; ++c) out[g * OUT_DIM + c] = lg[c] - m - lse;
}

static inline size_t align256(size_t x) { return (x + 255) & ~(size_t)255; }
static inline int cdiv(long long a, int b) { return (int)((a + b - 1) / b); }

extern "C" void kernel_launch(void* const* d_in, const int* in_sizes, int n_in,
                              void* d_out, int out_size, void* d_ws, size_t ws_size,
                              hipStream_t stream) {
    const float* x   = (const float*)d_in[0];
    const int*   ei  = (const int*)d_in[1];      // [2, E] flat
    const int*   bat = (const int*)d_in[2];
    const float* W1  = (const float*)d_in[3];
    const float* b1  = (const float*)d_in[4];
    const float* W2  = (const float*)d_in[5];
    const float* b2  = (const float*)d_in[6];
    const float* W3  = (const float*)d_in[7];
    const float* b3  = (const float*)d_in[8];
    const float* Wc  = (const float*)d_in[9];
    const float* bc  = (const float*)d_in[10];
    float* out = (float*)d_out;

    const int* src = ei;
    const int* dst = ei + N_EDGES;

    char* ws = (char*)d_ws;
    size_t o = 0;
    float* dinv   = (float*)(ws + o); o += align256((size_t)N_NODES * 4);
    float* enorm  = (float*)(ws + o); o += align256((size_t)N_EDGES * 4);
    float* buf0   = (float*)(ws + o); o += align256((size_t)N_NODES * HID * 4);
    float* buf1   = (float*)(ws + o); o += align256((size_t)N_NODES * HID * 4);
    float* pooled = (float*)(ws + o); o += align256((size_t)N_GRAPHS * HID2 * 4);
    float* counts = (float*)(ws + o); o += align256((size_t)N_GRAPHS * 4);
    (void)ws_size; (void)n_in; (void)in_sizes; (void)out_size;

    const int B = 256;

    // degrees -> dinv -> per-edge norms (reused by all 3 layers)
    k_fill1 <<<cdiv(N_NODES, B), B, 0, stream>>>(dinv, N_NODES);
    k_deg   <<<cdiv(N_EDGES, B), B, 0, stream>>>(dst, dinv, N_EDGES);
    k_rsqrt <<<cdiv(N_NODES, B), B, 0, stream>>>(dinv, N_NODES);
    k_norm  <<<cdiv(N_EDGES, B), B, 0, stream>>>(src, dst, dinv, enorm, N_EDGES);

    // ---- layer 1: 128 -> 64 ----
    k_gemm_wmma<<<N_NODES / 16, dim3(32, HID / 16), 0, stream>>>(x, W1, buf0, IN_DIM, HID);
    k_selfloop <<<cdiv((long long)N_NODES * (HID / 4), B), B, 0, stream>>>(buf0, dinv, buf1, HID);
    k_scatter  <<<cdiv((long long)N_EDGES * (HID / 4), B), B, 0, stream>>>(buf0, src, dst, enorm, buf1, HID);
    k_bias_relu<<<cdiv((long long)N_NODES * HID, B), B, 0, stream>>>(buf1, b1, buf0, HID);

    // ---- layer 2: 64 -> 64 ----
    k_gemm_wmma<<<N_NODES / 16, dim3(32, HID / 16), 0, stream>>>(buf0, W2, buf1, HID, HID);
    k_selfloop <<<cdiv((long long)N_NODES * (HID / 4), B), B, 0, stream>>>(buf1, dinv, buf0, HID);
    k_scatter  <<<cdiv((long long)N_EDGES * (HID / 4), B), B, 0, stream>>>(buf1, src, dst, enorm, buf0, HID);
    k_bias_relu<<<cdiv((long long)N_NODES * HID, B), B, 0, stream>>>(buf0, b2, buf1, HID);

    // ---- layer 3: 64 -> 32 ----
    k_gemm_wmma<<<N_NODES / 16, dim3(32, HID2 / 16), 0, stream>>>(buf1, W3, buf0, HID, HID2);
    k_selfloop <<<cdiv((long long)N_NODES * (HID2 / 4), B), B, 0, stream>>>(buf0, dinv, buf1, HID2);
    k_scatter  <<<cdiv((long long)N_EDGES * (HID2 / 4), B), B, 0, stream>>>(buf0, src, dst, enorm, buf1, HID2);
    k_bias_relu<<<cdiv((long long)N_NODES * HID2, B), B, 0, stream>>>(buf1, b3, buf0, HID2);

    // ---- pool + classify ----
    hipMemsetAsync(pooled, 0, (size_t)N_GRAPHS * HID2 * 4, stream);
    hipMemsetAsync(counts, 0, (size_t)N_GRAPHS * 4, stream);
    k_count<<<cdiv(N_NODES, B), B, 0, stream>>>(bat, counts, N_NODES);
    k_pool <<<cdiv((long long)N_NODES * (HID2 / 4), B), B, 0, stream>>>(buf0, bat, pooled);
    k_classifier<<<1, N_GRAPHS, 0, stream>>>(pooled, counts, Wc, bc, out);
}